// TotalLoss_54528904790213
// MI455X (gfx1250) — compile-verified
//
#include <hip/hip_runtime.h>
#include <hip/hip_bf16.h>
#include <math.h>

typedef __attribute__((ext_vector_type(2))) float v2f;
typedef __attribute__((ext_vector_type(8))) float v8f;

#define BATCH   32
#define NPTS    262144
#define BLOCKS_X 64           // blocks per batch
#define WAVES    8            // waves per block (256 threads)
#define PTS_PER_WAVE (NPTS / (BLOCKS_X * WAVES))   // 512
#define ITERS    (PTS_PER_WAVE / 16)               // 32 WMMAs per wave

// workspace layout (floats)
#define WS_TL_OFF 0            // transformation_loss scalar
#define WS_S_OFF  8            // S[b][c] accumulators: 32*4 floats
#define WS_M_OFF  144          // M matrices: 32*16 floats

// ---------------------------------------------------------------------------
// 4x4 inverse via adjugate (layout-agnostic: row-major in -> row-major out)
// ---------------------------------------------------------------------------
__device__ __forceinline__ void inv4(const float* m, float* out) {
    float inv[16];
    inv[0]  =  m[5]*m[10]*m[15] - m[5]*m[11]*m[14] - m[9]*m[6]*m[15] + m[9]*m[7]*m[14] + m[13]*m[6]*m[11] - m[13]*m[7]*m[10];
    inv[4]  = -m[4]*m[10]*m[15] + m[4]*m[11]*m[14] + m[8]*m[6]*m[15] - m[8]*m[7]*m[14] - m[12]*m[6]*m[11] + m[12]*m[7]*m[10];
    inv[8]  =  m[4]*m[9]*m[15]  - m[4]*m[11]*m[13] - m[8]*m[5]*m[15] + m[8]*m[7]*m[13] + m[12]*m[5]*m[11] - m[12]*m[7]*m[9];
    inv[12] = -m[4]*m[9]*m[14]  + m[4]*m[10]*m[13] + m[8]*m[5]*m[14] - m[8]*m[6]*m[13] - m[12]*m[5]*m[10] + m[12]*m[6]*m[9];
    inv[1]  = -m[1]*m[10]*m[15] + m[1]*m[11]*m[14] + m[9]*m[2]*m[15] - m[9]*m[3]*m[14] - m[13]*m[2]*m[11] + m[13]*m[3]*m[10];
    inv[5]  =  m[0]*m[10]*m[15] - m[0]*m[11]*m[14] - m[8]*m[2]*m[15] + m[8]*m[3]*m[14] + m[12]*m[2]*m[11] - m[12]*m[3]*m[10];
    inv[9]  = -m[0]*m[9]*m[15]  + m[0]*m[11]*m[13] + m[8]*m[1]*m[15] - m[8]*m[3]*m[13] - m[12]*m[1]*m[11] + m[12]*m[3]*m[9];
    inv[13] =  m[0]*m[9]*m[14]  - m[0]*m[10]*m[13] - m[8]*m[1]*m[14] + m[8]*m[2]*m[13] + m[12]*m[1]*m[10] - m[12]*m[2]*m[9];
    inv[2]  =  m[1]*m[6]*m[15]  - m[1]*m[7]*m[14]  - m[5]*m[2]*m[15] + m[5]*m[3]*m[14] + m[13]*m[2]*m[7]  - m[13]*m[3]*m[6];
    inv[6]  = -m[0]*m[6]*m[15]  + m[0]*m[7]*m[14]  + m[4]*m[2]*m[15] - m[4]*m[3]*m[14] - m[12]*m[2]*m[7]  + m[12]*m[3]*m[6];
    inv[10] =  m[0]*m[5]*m[15]  - m[0]*m[7]*m[13]  - m[4]*m[1]*m[15] + m[4]*m[3]*m[13] + m[12]*m[1]*m[7]  - m[12]*m[3]*m[5];
    inv[14] = -m[0]*m[5]*m[14]  + m[0]*m[6]*m[13]  + m[4]*m[1]*m[14] - m[4]*m[2]*m[13] - m[12]*m[1]*m[6]  + m[12]*m[2]*m[5];
    inv[3]  = -m[1]*m[6]*m[11]  + m[1]*m[7]*m[10]  + m[5]*m[2]*m[11] - m[5]*m[3]*m[10] - m[9]*m[2]*m[7]   + m[9]*m[3]*m[6];
    inv[7]  =  m[0]*m[6]*m[11]  - m[0]*m[7]*m[10]  - m[4]*m[2]*m[11] + m[4]*m[3]*m[10] + m[8]*m[2]*m[7]   - m[8]*m[3]*m[6];
    inv[11] = -m[0]*m[5]*m[11]  + m[0]*m[7]*m[9]   + m[4]*m[1]*m[11] - m[4]*m[3]*m[9]  - m[8]*m[1]*m[7]   + m[8]*m[3]*m[5];
    inv[15] =  m[0]*m[5]*m[10]  - m[0]*m[6]*m[9]   - m[4]*m[1]*m[10] + m[4]*m[2]*m[9]  + m[8]*m[1]*m[6]   - m[8]*m[2]*m[5];
    float det = m[0]*inv[0] + m[1]*inv[4] + m[2]*inv[8] + m[3]*inv[12];
    float id = 1.0f / det;
    #pragma unroll
    for (int i = 0; i < 16; ++i) out[i] = inv[i] * id;
}

// ---------------------------------------------------------------------------
// Setup: 1 block, 32 threads (one wave), thread b handles batch b.
// Computes transformation_loss, M[b] = inv(gt_rt[b]) @ RT_pred[b] - I,
// and zeroes the S accumulators.
// ---------------------------------------------------------------------------
__global__ __launch_bounds__(32) void setup_kernel(
    const float* __restrict__ gt_t, const float* __restrict__ gt_q,
    const float* __restrict__ pred_t, const float* __restrict__ pred_q,
    const float* __restrict__ gt_rt, float* __restrict__ ws)
{
    int b = threadIdx.x;  // 0..31

    float tt[3], pt[3];
    #pragma unroll
    for (int i = 0; i < 3; ++i) { tt[i] = gt_t[b*3+i]; pt[i] = pred_t[b*3+i]; }
    float tsq = 0.0f;
    #pragma unroll
    for (int i = 0; i < 3; ++i) { float d = pt[i] - tt[i]; tsq += d*d; }

    // normalized quaternions
    float q[4], r[4];
    {
        float s = 0.0f;
        #pragma unroll
        for (int i = 0; i < 4; ++i) { q[i] = pred_q[b*4+i]; s += q[i]*q[i]; }
        float inv = 1.0f / sqrtf(s);
        #pragma unroll
        for (int i = 0; i < 4; ++i) q[i] *= inv;
        s = 0.0f;
        #pragma unroll
        for (int i = 0; i < 4; ++i) { r[i] = gt_q[b*4+i]; s += r[i]*r[i]; }
        inv = 1.0f / sqrtf(s);
        #pragma unroll
        for (int i = 0; i < 4; ++i) r[i] *= inv;
    }

    // quaternion distance(q=pred, r=gt), ri = conj(r)
    float rw = r[0], rx = -r[1], ry = -r[2], rz = -r[3];
    float qw = q[0], qx = q[1], qy = q[2], qz = q[3];
    float w = rw*qw - rx*qx - ry*qy - rz*qz;
    float x = rw*qx + rx*qw - ry*qz + rz*qy;
    float y = rw*qy + rx*qz + ry*qw - rz*qx;
    float z = rw*qz - rx*qy + ry*qx + rz*qw;
    float dist = 2.0f * atan2f(sqrtf(x*x + y*y + z*z), fabsf(w));

    // wave reduction over 32 lanes
    float tsum = tsq, dsum = dist;
    #pragma unroll
    for (int off = 16; off > 0; off >>= 1) {
        tsum += __shfl_xor(tsum, off, 32);
        dsum += __shfl_xor(dsum, off, 32);
    }
    float loss_t = tsum / (float)(BATCH * 3);
    float loss_r = dsum / (float)BATCH;
    float tl = 2.0f * loss_t + 1.0f * loss_r;   // TRANS_W=2, ROT_W=1
    if (b == 0) ws[WS_TL_OFF] = tl;

    // RT_pred = T_pred @ R_pred  (row-major; R with translation in col 3)
    float R[16];
    {
        float w_ = qw, x_ = qx, y_ = qy, z_ = qz;
        R[0]  = 1.0f - 2.0f*y_*y_ - 2.0f*z_*z_;
        R[1]  = 2.0f*x_*y_ - 2.0f*z_*w_;
        R[2]  = 2.0f*x_*z_ + 2.0f*y_*w_;
        R[3]  = pt[0];
        R[4]  = 2.0f*x_*y_ + 2.0f*z_*w_;
        R[5]  = 1.0f - 2.0f*x_*x_ - 2.0f*z_*z_;
        R[6]  = 2.0f*y_*z_ - 2.0f*x_*w_;
        R[7]  = pt[1];
        R[8]  = 2.0f*x_*z_ - 2.0f*y_*w_;
        R[9]  = 2.0f*y_*z_ + 2.0f*x_*w_;
        R[10] = 1.0f - 2.0f*x_*x_ - 2.0f*y_*y_;
        R[11] = pt[2];
        R[12] = 0.0f; R[13] = 0.0f; R[14] = 0.0f; R[15] = 1.0f;
    }

    float G[16], Gi[16];
    #pragma unroll
    for (int i = 0; i < 16; ++i) G[i] = gt_rt[b*16 + i];
    inv4(G, Gi);

    // M = Gi @ RT_pred - I
    float M[16];
    #pragma unroll
    for (int i = 0; i < 4; ++i)
        #pragma unroll
        for (int j = 0; j < 4; ++j) {
            float acc = 0.0f;
            #pragma unroll
            for (int k = 0; k < 4; ++k) acc += Gi[i*4+k] * R[k*4+j];
            M[i*4+j] = acc - (i == j ? 1.0f : 0.0f);
        }

    float* Mws = ws + WS_M_OFF;
    #pragma unroll
    for (int i = 0; i < 16; ++i) Mws[b*16 + i] = M[i];

    float* S = ws + WS_S_OFF;
    #pragma unroll
    for (int c = 0; c < 4; ++c) S[b*4 + c] = 0.0f;
}

// ---------------------------------------------------------------------------
// Streaming WMMA kernel: grid (BLOCKS_X, BATCH), 256 threads (8 waves).
// Each wave: 32 x V_WMMA_F32_16X16X4_F32, 16 points per WMMA.
//   A (16x4): lane L (L<16) holds pc[pt][0:2], lane L+16 holds pc[pt][2:4]
//   B (4x16): col n<4 holds (RT_total - I)[:,n], cols 4..15 zero
//   D (16x16): col n holds diff for component n; square + accumulate.
// ---------------------------------------------------------------------------
__global__ __launch_bounds__(256) void pc_loss_kernel(
    const float* __restrict__ pc, const float* __restrict__ Mmat,
    float* __restrict__ S)
{
    const int b    = blockIdx.y;
    const int lane = threadIdx.x & 31;
    const int wave = threadIdx.x >> 5;
    const int n    = lane & 15;      // column / point-in-group index
    const int hi   = lane >> 4;      // 0: K=0,1  1: K=2,3

    // B matrix fragment (branchless: always load valid addr, mask to zero)
    const int   nn  = n & 3;
    const float msk = (n < 4) ? 1.0f : 0.0f;
    const float* Mb = Mmat + b * 16;
    v2f bv;
    bv[0] = Mb[(hi*2 + 0)*4 + nn] * msk;
    bv[1] = Mb[(hi*2 + 1)*4 + nn] * msk;

    const size_t base = (size_t)b * NPTS
                      + (size_t)blockIdx.x * (WAVES * PTS_PER_WAVE)
                      + (size_t)wave * PTS_PER_WAVE;
    const float* p0 = pc + (base + (size_t)n) * 4 + hi * 2;

    v8f acc = {};
    #pragma unroll 4
    for (int it = 0; it < ITERS; ++it) {
        v2f a = *(const v2f*)(p0 + (size_t)it * 64);   // next 16 points
        v8f c = {};
        v8f d = __builtin_amdgcn_wmma_f32_16x16x4_f32(
            false, a, false, bv, (short)0, c, false, false);
        acc += d * d;
    }

    float s = acc[0] + acc[1] + acc[2] + acc[3] + acc[4] + acc[5] + acc[6] + acc[7];
    s += __shfl_xor(s, 16, 32);      // fold M rows 0..7 with 8..15
    if (lane < 4) atomicAdd(&S[b*4 + lane], s);
}

// ---------------------------------------------------------------------------
// Finalize: sqrt over N-reduced sums, averages, final 3 scalars.
// ---------------------------------------------------------------------------
__global__ void finalize_kernel(const float* __restrict__ ws, float* __restrict__ out)
{
    if (threadIdx.x != 0 || blockIdx.x != 0) return;
    const float* S = ws + WS_S_OFF;
    float pcsum = 0.0f;
    for (int b = 0; b < BATCH; ++b) {
        float m = 0.0f;
        for (int c = 0; c < 4; ++c) m += sqrtf(S[b*4 + c]);
        pcsum += m * 0.25f;
    }
    float pc_mean = pcsum / (float)BATCH;
    float tl = ws[WS_TL_OFF];
    out[0] = 0.5f * tl + 0.5f * pc_mean;   // PC_W = 0.5
    out[1] = tl;
    out[2] = pc_mean;
}

extern "C" void kernel_launch(void* const* d_in, const int* in_sizes, int n_in,
                              void* d_out, int out_size, void* d_ws, size_t ws_size,
                              hipStream_t stream) {
    (void)in_sizes; (void)n_in; (void)out_size; (void)ws_size;
    const float* pc     = (const float*)d_in[0];
    const float* gt_t   = (const float*)d_in[1];
    const float* gt_q   = (const float*)d_in[2];
    const float* pred_t = (const float*)d_in[3];
    const float* pred_q = (const float*)d_in[4];
    const float* gt_rt  = (const float*)d_in[5];
    float* out = (float*)d_out;
    float* ws  = (float*)d_ws;

    setup_kernel<<<1, 32, 0, stream>>>(gt_t, gt_q, pred_t, pred_q, gt_rt, ws);
    dim3 grid(BLOCKS_X, BATCH);
    pc_loss_kernel<<<grid, 256, 0, stream>>>(pc, ws + WS_M_OFF, ws + WS_S_OFF);
    finalize_kernel<<<1, 32, 0, stream>>>(ws, out);
}